// MonotoneActivation_19524921328060
// MI455X (gfx1250) — compile-verified
//
#include <hip/hip_runtime.h>
#include <cstdint>

#define GROUPS 256
#define TABLE  16
#define ROWS_PER_BLOCK 16

typedef float v4f __attribute__((ext_vector_type(4)));

// One thread per (batch row, group). Block = 256 threads = 8 wave32, one group
// per thread, ROWS_PER_BLOCK batch rows per block. Params (64 KB) staged into
// LDS once per block with CDNA5 async global->LDS DMA, XOR-swizzled to avoid
// LDS bank conflicts on the per-lane corner gathers.
__global__ __launch_bounds__(256) void lattice_simplex_kernel(
    const v4f* __restrict__ Xv,   // [B * 256] float4 (one per (b,g))
    const v4f* __restrict__ Pv,   // [256 * 16] float4 (one per (g,t))
    v4f* __restrict__ Ov,         // [B * 256] float4
    int batch)
{
    __shared__ v4f lut[GROUPS * TABLE];   // 64 KB

    const int tid = threadIdx.x;             // 0..255 == group id

    // ---- Stage params into LDS via async global->LDS (ASYNCcnt path) ----
    #pragma unroll
    for (int it = 0; it < (GROUPS * TABLE) / 256; ++it) {
        int id = it * 256 + tid;             // flat (g,t)
        int g  = id >> 4;
        int t  = id & 15;
        int slot = (g << 4) | (t ^ (g & 15));            // XOR bank swizzle
        unsigned int ldsOff =
            (unsigned int)(uintptr_t)(&lut[slot]);       // LDS byte address
        unsigned long long ga =
            (unsigned long long)(uintptr_t)(Pv + id);    // global byte address
        asm volatile("global_load_async_to_lds_b128 %0, %1, off"
                     :: "v"(ldsOff), "v"(ga)
                     : "memory");
    }
    asm volatile("s_wait_asynccnt 0" ::: "memory");
    __syncthreads();

    const int g     = tid;
    const int gsw   = g & 15;
    const int gbase = g << 4;
    const v4f p15 = lut[gbase | (15 ^ gsw)];             // idx[0] == 15 always

    const int row0 = blockIdx.x * ROWS_PER_BLOCK;

    for (int r = 0; r < ROWS_PER_BLOCK; ++r) {
        const int b = row0 + r;
        const v4f* xp = Xv + (size_t)b * GROUPS + g;
        if (r + 1 < ROWS_PER_BLOCK)
            __builtin_prefetch(xp + GROUPS, 0, 0);       // global_prefetch_b8

        v4f x = __builtin_nontemporal_load(xp);
        float x0 = x.x, x1 = x.y, x2 = x.z, x3 = x.w;

        // Stable ranks: rank_i = #{j : x_j < x_i or (x_j == x_i and j < i)}
        int n01 = x0 > x1;
        int n02 = x0 > x2;
        int n03 = x0 > x3;
        int n12 = x1 > x2;
        int n13 = x1 > x3;
        int n23 = x2 > x3;
        int r0v = n01 + n02 + n03;
        int r1v = (1 - n01) + n12 + n13;
        int r2v = (2 - n02 - n12) + n23;
        int r3v = 3 - n03 - n13 - n23;

        // Sorted values via 5-exchange min/max network
        float lo01 = fminf(x0, x1), hi01 = fmaxf(x0, x1);
        float lo23 = fminf(x2, x3), hi23 = fmaxf(x2, x3);
        float s0 = fminf(lo01, lo23);
        float ta = fmaxf(lo01, lo23);
        float s3 = fmaxf(hi01, hi23);
        float tb = fminf(hi01, hi23);
        float s1 = fminf(ta, tb);
        float s2 = fmaxf(ta, tb);

        // Lattice masks: m_k = bits of elements with rank >= k (m_0 == 15)
        int m1 = (r0v > 0 ? 1 : 0) | (r1v > 0 ? 2 : 0) |
                 (r2v > 0 ? 4 : 0) | (r3v > 0 ? 8 : 0);
        int m2 = (r0v > 1 ? 1 : 0) | (r1v > 1 ? 2 : 0) |
                 (r2v > 1 ? 4 : 0) | (r3v > 1 ? 8 : 0);
        int m3 = (r0v > 2 ? 1 : 0) | (r1v > 2 ? 2 : 0) |
                 (r2v > 2 ? 4 : 0) | (r3v > 2 ? 8 : 0);

        // Gather lattice corners from swizzled LDS
        v4f q1 = lut[gbase | (m1 ^ gsw)];
        v4f q2 = lut[gbase | (m2 ^ gsw)];
        v4f q3 = lut[gbase | (m3 ^ gsw)];

        float c1 = s1 - s0;
        float c2 = s2 - s1;
        float c3 = s3 - s2;

        v4f o;
        o.x = fmaf(s0, p15.x, fmaf(c1, q1.x, fmaf(c2, q2.x, c3 * q3.x)));
        o.y = fmaf(s0, p15.y, fmaf(c1, q1.y, fmaf(c2, q2.y, c3 * q3.y)));
        o.z = fmaf(s0, p15.z, fmaf(c1, q1.z, fmaf(c2, q2.z, c3 * q3.z)));
        o.w = fmaf(s0, p15.w, fmaf(c1, q1.w, fmaf(c2, q2.w, c3 * q3.w)));

        __builtin_nontemporal_store(o, Ov + (size_t)b * GROUPS + g);
    }
}

extern "C" void kernel_launch(void* const* d_in, const int* in_sizes, int n_in,
                              void* d_out, int out_size, void* d_ws, size_t ws_size,
                              hipStream_t stream) {
    const v4f* X = (const v4f*)d_in[0];   // [B, GROUPS*4] f32
    const v4f* P = (const v4f*)d_in[1];   // [GROUPS, 16, 4] f32
    v4f* O = (v4f*)d_out;                 // [B, GROUPS*4] f32

    const int batch  = in_sizes[0] / (GROUPS * 4);     // 16384
    const int blocks = batch / ROWS_PER_BLOCK;         // 1024

    lattice_simplex_kernel<<<blocks, 256, 0, stream>>>(X, P, O, batch);
}